// Local_Grouper_21397527069041
// MI455X (gfx1250) — compile-verified
//
#include <hip/hip_runtime.h>
#include <math.h>
#include <stdint.h>

// ---- problem constants -----------------------------------------------------
#define BATCH   16
#define NPTS    4096
#define GROUPS  512          // GROUP_NUM
#define KNN     32           // K_NEIGHBORS
#define CH      64           // feature channels
#define NTILES  (NPTS / 16)  // 256 WMMA N-tiles per query tile
#define QTILES  (BATCH * (GROUPS / 16))   // 512 16-query tiles

typedef __attribute__((ext_vector_type(2))) float v2f;
typedef __attribute__((ext_vector_type(8))) float v8f;

// ---------------------------------------------------------------------------
// K1: farthest point sampling. One block per batch, 256 threads, each thread
// owns 16 points in registers. 511 sequential iterations, LDS tree argmax.
// Strict '>' keeps the lowest index on ties (matches jnp.argmax).
// ---------------------------------------------------------------------------
__global__ __launch_bounds__(256) void fps_kernel(const float* __restrict__ xyz,
                                                  int* __restrict__ fps_idx) {
    const int b   = blockIdx.x;
    const int tid = threadIdx.x;
    const float* px = xyz + (size_t)b * NPTS * 3;

    float cx[16], cy[16], cz[16], dist[16];
#pragma unroll
    for (int i = 0; i < 16; ++i) {
        const int p = tid * 16 + i;
        cx[i] = px[p * 3 + 0];
        cy[i] = px[p * 3 + 1];
        cz[i] = px[p * 3 + 2];
        dist[i] = 1.0e10f;
    }

    __shared__ float s_val[256];
    __shared__ int   s_idx[256];
    __shared__ float s_cent[3];

    int last = 0;
    if (tid == 0) fps_idx[b * GROUPS + 0] = 0;

    for (int it = 1; it < GROUPS; ++it) {
        if ((last >> 4) == tid) {               // owner broadcasts centroid
            const int j = last & 15;
            s_cent[0] = cx[j]; s_cent[1] = cy[j]; s_cent[2] = cz[j];
        }
        __syncthreads();
        const float mx = s_cent[0], my = s_cent[1], mz = s_cent[2];

        float best = -1.0f; int bidx = tid * 16;
#pragma unroll
        for (int i = 0; i < 16; ++i) {
            const float dx = cx[i] - mx, dy = cy[i] - my, dz = cz[i] - mz;
            const float d  = dx * dx + dy * dy + dz * dz;
            dist[i] = fminf(dist[i], d);
            if (dist[i] > best) { best = dist[i]; bidx = tid * 16 + i; }
        }
        s_val[tid] = best; s_idx[tid] = bidx;
        __syncthreads();
        for (int s = 128; s > 0; s >>= 1) {
            if (tid < s && s_val[tid + s] > s_val[tid]) {
                s_val[tid] = s_val[tid + s]; s_idx[tid] = s_idx[tid + s];
            }
            __syncthreads();
        }
        last = s_idx[0];
        if (tid == 0) fps_idx[b * GROUPS + it] = last;
        __syncthreads();   // everyone consumed s_idx[0] before next iteration
    }
}

// ---------------------------------------------------------------------------
// K2: gather lc_xyz (B,G,3) and lc_x (B,G,64). One block per (b,g).
// ---------------------------------------------------------------------------
__global__ __launch_bounds__(64) void gather_lc_kernel(const float* __restrict__ xyz,
                                                       const float* __restrict__ x,
                                                       const int* __restrict__ fps_idx,
                                                       float* __restrict__ lc_xyz,
                                                       float* __restrict__ lc_x) {
    const int bg = blockIdx.x;
    const int b  = bg >> 9;
    const int t  = threadIdx.x;
    const int idx = fps_idx[bg];
    lc_x[(size_t)bg * CH + t] = x[((size_t)b * NPTS + idx) * CH + t];
    if (t < 3) lc_xyz[(size_t)bg * 3 + t] = xyz[((size_t)b * NPTS + idx) * 3 + t];
}

// ---------------------------------------------------------------------------
// K3: KNN ranking keys via V_WMMA_F32_16X16X4_F32.
//   D'[g][n] = p_n^2 - 2 q_g . p_n   (row-shift of true distance: same top-k)
//   A row m = [-2qx, -2qy, -2qz, 1]   B col n = [px, py, pz, p2]^T
// One wave per 16-query tile, looping 256 N-tiles. 4 waves / block.
// Results staged into the knn_x output slot (reused as scratch; fits in L2).
// ---------------------------------------------------------------------------
__global__ __launch_bounds__(128) void knn_dist_kernel(const float* __restrict__ xyz,
                                                       const float* __restrict__ lc_xyz,
                                                       float* __restrict__ Dscr) {
    const int wave  = threadIdx.x >> 5;
    const int lane  = threadIdx.x & 31;
    const int T     = blockIdx.x * 4 + wave;     // query tile id [0,512)
    const int b     = T >> 5;
    const int gbase = (T & 31) * 16;
    const int m     = lane & 15;                 // row / column within tile
    const int half  = lane >> 4;

    // A fragment (loop invariant): lanes 0-15 hold K={0,1}, lanes 16-31 K={2,3}
    const float* q = lc_xyz + ((size_t)(b * GROUPS + gbase + m)) * 3;
    const float qx = q[0], qy = q[1], qz = q[2];
    v2f a;
    a.x = half ? (-2.0f * qz) : (-2.0f * qx);
    a.y = half ? 1.0f         : (-2.0f * qy);

    const float* pbase = xyz + (size_t)b * NPTS * 3;
    const size_t rowb  = ((size_t)(b * GROUPS + gbase + 8 * half)) * NPTS;

#pragma unroll 2
    for (int nt = 0; nt < NTILES; ++nt) {
        const int n = nt * 16 + m;
        // prefetch one N-tile ahead (global_prefetch_b8)
        __builtin_prefetch(pbase + (size_t)(n + 16) * 3, 0, 3);
        const float px = pbase[n * 3 + 0];
        const float py = pbase[n * 3 + 1];
        const float pz = pbase[n * 3 + 2];
        const float p2 = px * px + py * py + pz * pz;

        v2f bf;                                   // B fragment, mirrors A layout
        bf.x = half ? pz : px;
        bf.y = half ? p2 : py;

        v8f c = {};
        c = __builtin_amdgcn_wmma_f32_16x16x4_f32(
                /*neg_a=*/false, a, /*neg_b=*/false, bf,
                /*c_mod=*/(short)0, c, /*reuse_a=*/false, /*reuse_b=*/false);

        float* dst = Dscr + rowb + (size_t)n;     // column n, rows r(+8*half)
#pragma unroll
        for (int r = 0; r < 8; ++r)
            dst[(size_t)r * NPTS] = c[r];
    }
}

// ---------------------------------------------------------------------------
// K4: top-32 smallest per query row. Block per (b,g). The 16 KB row is staged
// global -> LDS with CDNA5 async copies (GLOBAL_LOAD_ASYNC_TO_LDS_B128,
// ASYNCcnt-tracked, no VGPR staging), then 32 masked argmin passes.
// Strict '<' => lowest index wins ties (matches lax.top_k order).
// ---------------------------------------------------------------------------
__global__ __launch_bounds__(128) void topk_kernel(const float* __restrict__ Dscr,
                                                   int* __restrict__ knn_idx) {
    __shared__ float row[NPTS];
    __shared__ float rv[128];
    __shared__ int   ri[128];
    const int bg = blockIdx.x;
    const int t  = threadIdx.x;
    const float* src = Dscr + (size_t)bg * NPTS;

    // async stage: 1024 x 16B chunks, 8 per lane, direct to LDS
    {
        const uint32_t lds_base = (uint32_t)(uintptr_t)(&row[0]);
        const uint64_t g_base   = (uint64_t)(uintptr_t)src;
#pragma unroll
        for (int c = 0; c < (NPTS / 4) / 128; ++c) {
            const int i = c * 128 + t;
            const uint32_t ldsa = lds_base + (uint32_t)i * 16u;
            const uint64_t ga   = g_base + (uint64_t)i * 16u;
            asm volatile("global_load_async_to_lds_b128 %0, %1, off"
                         :: "v"(ldsa), "v"(ga) : "memory");
        }
        asm volatile("s_wait_asynccnt 0x0" ::: "memory");
    }
    __syncthreads();

    for (int k = 0; k < KNN; ++k) {
        float best = 3.0e38f; int bidx = t * 32;
#pragma unroll 4
        for (int j = 0; j < 32; ++j) {
            const float v = row[t * 32 + j];
            if (v < best) { best = v; bidx = t * 32 + j; }
        }
        rv[t] = best; ri[t] = bidx;
        __syncthreads();
        for (int s = 64; s > 0; s >>= 1) {
            if (t < s && rv[t + s] < rv[t]) { rv[t] = rv[t + s]; ri[t] = ri[t + s]; }
            __syncthreads();
        }
        if (t == 0) {
            const int sel = ri[0];
            knn_idx[(size_t)bg * KNN + k] = sel;
            row[sel] = 3.0e38f;                   // mask out
        }
        __syncthreads();
    }
}

// ---------------------------------------------------------------------------
// K5: diff = gather(xyz, knn_idx) - lc_xyz; write un-normalized knn_xyz and
// per-block partial sum / sumsq (96 elements per (b,g)).
// ---------------------------------------------------------------------------
__global__ __launch_bounds__(96) void diff_kernel(const float* __restrict__ xyz,
                                                  const float* __restrict__ lc_xyz,
                                                  const int* __restrict__ knn_idx,
                                                  float* __restrict__ knn_xyz,
                                                  float* __restrict__ psum,
                                                  float* __restrict__ psumsq) {
    __shared__ float sv[96], sq[96];
    const int bg = blockIdx.x;
    const int b  = bg >> 9;
    const int t  = threadIdx.x;
    const int k  = t / 3, c = t % 3;
    const int nidx = knn_idx[(size_t)bg * KNN + k];
    const float d = xyz[((size_t)b * NPTS + nidx) * 3 + c]
                  - lc_xyz[(size_t)bg * 3 + c];
    knn_xyz[(size_t)bg * (KNN * 3) + t] = d;
    sv[t] = d; sq[t] = d * d;
    __syncthreads();
    for (int s = 48; s >= 3; s >>= 1) {
        if (t < s) { sv[t] += sv[t + s]; sq[t] += sq[t + s]; }
        __syncthreads();
    }
    if (t == 0) { psum[bg] = sv[0] + sv[1] + sv[2]; psumsq[bg] = sq[0] + sq[1] + sq[2]; }
}

// K6: reduce 8192 partials -> 1/(std+1e-5), ddof=1 over 786432 elements.
__global__ __launch_bounds__(256) void std_kernel(const float* __restrict__ psum,
                                                  const float* __restrict__ psumsq,
                                                  float* __restrict__ inv_std) {
    __shared__ float sv[256], sq[256];
    const int t = threadIdx.x;
    float a = 0.0f, bb = 0.0f;
    for (int i = t; i < BATCH * GROUPS; i += 256) { a += psum[i]; bb += psumsq[i]; }
    sv[t] = a; sq[t] = bb;
    __syncthreads();
    for (int s = 128; s > 0; s >>= 1) {
        if (t < s) { sv[t] += sv[t + s]; sq[t] += sq[t + s]; }
        __syncthreads();
    }
    if (t == 0) {
        const double n = (double)(BATCH * GROUPS * KNN * 3);
        const double s1 = (double)sv[0], s2 = (double)sq[0];
        double var = (s2 - s1 * s1 / n) / (n - 1.0);
        if (var < 0.0) var = 0.0;
        inv_std[0] = 1.0f / ((float)sqrt(var) + 1e-5f);
    }
}

// K7: scale knn_xyz in place by 1/(std+1e-5).
__global__ __launch_bounds__(256) void norm_kernel(float* __restrict__ knn_xyz,
                                                   const float* __restrict__ inv_std) {
    const float s = inv_std[0];
    const size_t i = (size_t)blockIdx.x * 256 + threadIdx.x;
    knn_xyz[i] *= s;
}

// K8: knn_x = concat(gather(x, knn_idx), broadcast lc_x). Overwrites scratch.
__global__ __launch_bounds__(128) void gather_knnx_kernel(const float* __restrict__ x,
                                                          const float* __restrict__ lc_x,
                                                          const int* __restrict__ knn_idx,
                                                          float* __restrict__ knn_x) {
    __shared__ int s_idx[KNN];
    const int bg = blockIdx.x;
    const int b  = bg >> 9;
    const int t  = threadIdx.x;
    if (t < KNN) s_idx[t] = knn_idx[(size_t)bg * KNN + t];
    const float lcv = (t >= CH) ? lc_x[(size_t)bg * CH + (t - CH)] : 0.0f;
    __syncthreads();
    for (int k = 0; k < KNN; ++k) {
        const int nidx = s_idx[k];
        const float v = (t < CH) ? x[((size_t)b * NPTS + nidx) * CH + t] : lcv;
        knn_x[((size_t)bg * KNN + k) * (2 * CH) + t] = v;
    }
}

// ---------------------------------------------------------------------------
extern "C" void kernel_launch(void* const* d_in, const int* in_sizes, int n_in,
                              void* d_out, int out_size, void* d_ws, size_t ws_size,
                              hipStream_t stream) {
    const float* xyz = (const float*)d_in[0];   // (16,4096,3)
    const float* x   = (const float*)d_in[1];   // (16,4096,64)

    // flat output layout: lc_xyz | lc_x | knn_xyz | knn_x
    float* out     = (float*)d_out;
    float* lc_xyz  = out;                                   // 16*512*3      = 24576
    float* lc_x    = out + 24576;                           // 16*512*64     = 524288
    float* knn_xyz = out + 24576 + 524288;                  // 16*512*32*3   = 786432
    float* knn_x   = out + 24576 + 524288 + 786432;         // 16*512*32*128 = 33554432
    float* Dscr    = knn_x;   // distance-key scratch lives in the knn_x slot

    // small workspace (~1.1 MB)
    int*   fps_idx = (int*)d_ws;                            // 8192
    int*   knn_idx = fps_idx + BATCH * GROUPS;              // 262144
    float* psum    = (float*)(knn_idx + BATCH * GROUPS * KNN);
    float* psumsq  = psum + BATCH * GROUPS;
    float* inv_std = psumsq + BATCH * GROUPS;

    fps_kernel      <<<BATCH,          256, 0, stream>>>(xyz, fps_idx);
    gather_lc_kernel<<<BATCH * GROUPS,  64, 0, stream>>>(xyz, x, fps_idx, lc_xyz, lc_x);
    knn_dist_kernel <<<QTILES / 4,     128, 0, stream>>>(xyz, lc_xyz, Dscr);
    topk_kernel     <<<BATCH * GROUPS, 128, 0, stream>>>(Dscr, knn_idx);
    diff_kernel     <<<BATCH * GROUPS,  96, 0, stream>>>(xyz, lc_xyz, knn_idx,
                                                         knn_xyz, psum, psumsq);
    std_kernel      <<<1,              256, 0, stream>>>(psum, psumsq, inv_std);
    norm_kernel     <<<(BATCH * GROUPS * KNN * 3) / 256, 256, 0, stream>>>(knn_xyz, inv_std);
    gather_knnx_kernel<<<BATCH * GROUPS, 128, 0, stream>>>(x, lc_x, knn_idx, knn_x);
}